// DynamicScaling_283467842320
// MI455X (gfx1250) — compile-verified
//
#include <hip/hip_runtime.h>

// out[r, c] = x[r, c] * weight[c] + bias[c]
// N = 16384 rows, H = 4096 cols, fp32. Pure streaming: ~512 MiB HBM traffic,
// 0.25 FLOP/byte -> bandwidth-bound. Optimize for 23.3 TB/s, not FLOPs.

typedef float v4f __attribute__((ext_vector_type(4)));

#define HDIM   4096
#define HDIM4  (HDIM / 4)   // 1024 float4 per row
#define TPB    256          // 8 wave32 per block
#define COLS_PER_THREAD (HDIM4 / TPB)  // 4 float4 per thread per row

__global__ __launch_bounds__(TPB)
void scale_shift_f32_kernel(const float* __restrict__ x,
                            const float* __restrict__ weight,
                            const float* __restrict__ bias,
                            float* __restrict__ out,
                            int nrows)
{
    // Stage weight/bias in LDS: 2 * 16 KiB = 32 KiB of the 320 KiB WGP pool.
    // These are re-read every row; serving them from LDS (ds_load_b128) keeps
    // the vector-memory pipe free for the two big NT streams.
    __shared__ v4f s_w[HDIM4];
    __shared__ v4f s_b[HDIM4];

    const v4f* __restrict__ w4 = (const v4f*)weight;
    const v4f* __restrict__ b4 = (const v4f*)bias;
    for (int i = threadIdx.x; i < HDIM4; i += TPB) {
        s_w[i] = w4[i];          // global_load_b128 (RT: tiny, cache-friendly)
        s_b[i] = b4[i];          // ds_store_b128
    }
    __syncthreads();

    const v4f* __restrict__ x4 = (const v4f*)x;
    v4f*       __restrict__ o4 = (v4f*)out;

    const int    tid  = threadIdx.x;
    const size_t rstep = (size_t)gridDim.x * HDIM4;

    // Grid-stride over rows; each block sweeps a full 16 KiB row per iteration.
    for (int r = blockIdx.x; r < nrows; r += gridDim.x) {
        const v4f* __restrict__ xr = x4 + (size_t)r * HDIM4;
        v4f*       __restrict__ orow = o4 + (size_t)r * HDIM4;

        // Prefetch this block's NEXT row while streaming the current one.
        // 256 lanes x 1 byte each -> 256 distinct cachelines cover the 16 KiB
        // row exactly; lowers to global_prefetch_b8 (speculative, fault-safe).
        if (r + (int)gridDim.x < nrows) {
            __builtin_prefetch((const char*)(xr + rstep) + (size_t)tid * 64, 0, 0);
        }

#pragma unroll
        for (int k = 0; k < COLS_PER_THREAD; ++k) {
            const int c = tid + k * TPB;
            // x: read-once stream larger than L2 -> non-temporal load (TH=NT)
            v4f xv = __builtin_nontemporal_load(&xr[c]);
            v4f wv = s_w[c];     // ds_load_b128
            v4f bv = s_b[c];     // ds_load_b128
            v4f res = xv * wv + bv;   // v_fma (contracted at -O3)
            // out: write-once stream -> non-temporal store (TH=NT)
            __builtin_nontemporal_store(res, &orow[c]);
        }
    }
}

extern "C" void kernel_launch(void* const* d_in, const int* in_sizes, int n_in,
                              void* d_out, int out_size, void* d_ws, size_t ws_size,
                              hipStream_t stream)
{
    const float* x      = (const float*)d_in[0];
    const float* weight = (const float*)d_in[1];
    const float* bias   = (const float*)d_in[2];
    float*       out    = (float*)d_out;

    const int nrows = in_sizes[0] / HDIM;   // 16384

    // Enough blocks to fill every WGP several times over (8 waves each) while
    // letting each block amortize its 32 KiB LDS staging across ~8 rows.
    int grid = nrows / 8;
    if (grid < 1) grid = 1;

    scale_shift_f32_kernel<<<grid, TPB, 0, stream>>>(x, weight, bias, out, nrows);
}